// MSA_block_76562087018873
// MI455X (gfx1250) — compile-verified
//
#include <hip/hip_runtime.h>
#include <hip/hip_bf16.h>

typedef float v2f __attribute__((ext_vector_type(2)));
typedef float v8f __attribute__((ext_vector_type(8)));

#define BATCH   64
#define NTOK    196
#define NPAD    208
#define EMBEDC  128
#define NHEADS  32
#define ROWS    (BATCH * NTOK)        // 12544
#define MTILES  (ROWS / 16)           // 784
#define AST     212                   // padded LDS row stride (floats) for attn strip

static __device__ __forceinline__ v8f wmma_f32_4(v2f a, v2f b, v8f c) {
    return __builtin_amdgcn_wmma_f32_16x16x4_f32(false, a, false, b, (short)0, c,
                                                 false, false);
}

// Async global->LDS B128 copy (CDNA5, tracked by ASYNCcnt).
static __device__ __forceinline__ void async_copy_b128(void* lds_dst,
                                                       const void* gsrc) {
    unsigned lds = (unsigned)(unsigned long long)lds_dst; // addrspace(3) offset
    asm volatile("global_load_async_to_lds_b128 %0, %1, off"
                 :: "v"(lds), "v"(gsrc) : "memory");
}
static __device__ __forceinline__ void wait_asynccnt0() {
    asm volatile("s_wait_asynccnt 0x0" ::: "memory");
}

// ---------------------------------------------------------------------------
// Kernel 1: QKV projection GEMM for both branches.
//   [12544 x 128] @ qkv_w^T [128 x 384] using V_WMMA_F32_16X16X4_F32.
//   Writes q~ [b,h,n,8] (branch-concat, pre-scaled by 0.5), k~ [b,h,n,8],
//   vcat [b, 2H, n*4].
// ---------------------------------------------------------------------------
__global__ __launch_bounds__(256) void qkv_kernel(
    const float* __restrict__ x1, const float* __restrict__ x2,
    const float* __restrict__ w,
    float* __restrict__ qb, float* __restrict__ kb, float* __restrict__ vcat)
{
    const int mt = blockIdx.x;      // 0..783
    const int br = blockIdx.y;      // branch 0/1
    const float* __restrict__ x = br ? x2 : x1;

    __shared__ float xs[16 * 128];
    const int t = threadIdx.x;
    // async stage: 16 rows x 128 floats = 512 x 16B
    for (int i = t; i < 512; i += 256)
        async_copy_b128(&xs[i * 4], x + (size_t)mt * 16 * 128 + i * 4);
    wait_asynccnt0();
    __syncthreads();

    const int lane = t & 31;
    const int wave = t >> 5;
    const int half = lane >> 4;     // 0: K={0,1}, 1: K={2,3}
    const int l16  = lane & 15;
    const int koff = half * 2;

    for (int nt = wave * 3; nt < wave * 3 + 3; ++nt) {
        v8f c = {};
        const float* wrow = w + (nt * 16 + l16) * 128;   // B[k,n] = qkv_w[n,k]
        #pragma unroll
        for (int kk = 0; kk < 32; ++kk) {
            v2f a, b;
            a.x = xs[l16 * 128 + kk * 4 + koff];
            a.y = xs[l16 * 128 + kk * 4 + koff + 1];
            b.x = wrow[kk * 4 + koff];
            b.y = wrow[kk * 4 + koff + 1];
            c = wmma_f32_4(a, b, c);
        }
        const int o = nt * 16 + l16;
        #pragma unroll
        for (int r = 0; r < 8; ++r) {
            int m    = r + half * 8;
            int grow = mt * 16 + m;
            int bi   = grow / NTOK;
            int n    = grow - bi * NTOK;
            float val = c[r];
            if (o < 128) {
                int h = o >> 2, d = o & 3;
                qb[((bi * NHEADS + h) * NTOK + n) * 8 + br * 4 + d] = val * 0.5f;
            } else if (o < 256) {
                int oo = o - 128, h = oo >> 2, d = oo & 3;
                kb[((bi * NHEADS + h) * NTOK + n) * 8 + br * 4 + d] = val;
            } else {
                int oo = o - 256, h = oo >> 2, d = oo & 3;
                vcat[(bi * 64 + br * 32 + h) * (NTOK * 4) + n * 4 + d] = val;
            }
        }
    }
}

// ---------------------------------------------------------------------------
// Kernel 2: value fusion (1x1 conv 64->32) + InstanceNorm + relu + sigmoid.
// One block per batch window; two passes (stats then normalize).
// ---------------------------------------------------------------------------
__global__ __launch_bounds__(256) void fuse_kernel(
    const float* __restrict__ vcat, const float* __restrict__ fw,
    const float* __restrict__ fb, float* __restrict__ vf)
{
    const int b = blockIdx.x;
    __shared__ float sw[32 * 64];
    __shared__ float sred[2 * 32 * 8];   // [sum/ssq][o][wave]
    __shared__ float smv[64];            // mu[32], inv[32]
    const int t = threadIdx.x;
    // fuse_w is 8 KB = 512 x 16B: async stage
    for (int i = t; i < 512; i += 256) async_copy_b128(&sw[i * 4], fw + i * 4);
    wait_asynccnt0();
    __syncthreads();

    const int lane = t & 31, wave = t >> 5;
    float sum[32], ssq[32];
    #pragma unroll
    for (int o = 0; o < 32; ++o) { sum[o] = 0.f; ssq[o] = 0.f; }

    for (int p = t; p < NTOK * 4; p += 256) {
        float acc[32];
        #pragma unroll
        for (int o = 0; o < 32; ++o) acc[o] = fb[o];
        for (int c = 0; c < 64; ++c) {
            float xc = vcat[(b * 64 + c) * (NTOK * 4) + p];
            #pragma unroll
            for (int o = 0; o < 32; ++o) acc[o] = fmaf(sw[o * 64 + c], xc, acc[o]);
        }
        #pragma unroll
        for (int o = 0; o < 32; ++o) { sum[o] += acc[o]; ssq[o] += acc[o] * acc[o]; }
    }
    #pragma unroll
    for (int o = 0; o < 32; ++o) {
        float s = sum[o], q = ssq[o];
        for (int d = 16; d >= 1; d >>= 1) {
            s += __shfl_xor(s, d, 32);
            q += __shfl_xor(q, d, 32);
        }
        if (lane == 0) { sred[o * 8 + wave] = s; sred[256 + o * 8 + wave] = q; }
    }
    __syncthreads();
    if (t < 32) {
        float s = 0.f, q = 0.f;
        for (int wv = 0; wv < 8; ++wv) { s += sred[t * 8 + wv]; q += sred[256 + t * 8 + wv]; }
        float mu  = s * (1.0f / 784.0f);
        float var = q * (1.0f / 784.0f) - mu * mu;
        smv[t]      = mu;
        smv[32 + t] = rsqrtf(var + 1e-5f);
    }
    __syncthreads();

    for (int p = t; p < NTOK * 4; p += 256) {
        float acc[32];
        #pragma unroll
        for (int o = 0; o < 32; ++o) acc[o] = fb[o];
        for (int c = 0; c < 64; ++c) {
            float xc = vcat[(b * 64 + c) * (NTOK * 4) + p];
            #pragma unroll
            for (int o = 0; o < 32; ++o) acc[o] = fmaf(sw[o * 64 + c], xc, acc[o]);
        }
        #pragma unroll
        for (int o = 0; o < 32; ++o) {
            float v = (acc[o] - smv[o]) * smv[32 + o];
            v = fmaxf(v, 0.f);
            v = 1.0f / (1.0f + __expf(-v));
            vf[(b * 32 + o) * (NTOK * 4) + p] = v;
        }
    }
}

// ---------------------------------------------------------------------------
// Kernel 3: expand relative-position bias to padded [H, 208, 208] with
// -1e30 masks on padded rows/cols (drives softmax masking for free).
// ---------------------------------------------------------------------------
__global__ __launch_bounds__(256) void bias_kernel(
    const float* __restrict__ tbl, const int* __restrict__ rel,
    float* __restrict__ bias)
{
    int idx = blockIdx.x * 256 + threadIdx.x;
    if (idx >= NHEADS * NPAD * NPAD) return;
    int h = idx / (NPAD * NPAD);
    int r = idx - h * NPAD * NPAD;
    int n = r / NPAD, m = r - n * NPAD;
    float v = -1e30f;
    if (n < NTOK && m < NTOK) v = tbl[rel[n * NTOK + m] * NHEADS + h];
    bias[idx] = v;
}

// ---------------------------------------------------------------------------
// Kernel 4: fused attention per (b,h). Scores via 2x WMMA (dual-branch K=8),
// softmax with half-wave shfl reductions, attn@v via WMMA through an LDS strip.
// q~/k~/v staged into LDS with GLOBAL_LOAD_ASYNC_TO_LDS_B128.
// ---------------------------------------------------------------------------
__global__ __launch_bounds__(256) void attn_kernel(
    const float* __restrict__ qb, const float* __restrict__ kb,
    const float* __restrict__ vf, const float* __restrict__ bias,
    float* __restrict__ xa)
{
    extern __shared__ float sm[];
    float* s_q = sm;                    // 208*8
    float* s_k = sm + NPAD * 8;        // 208*8
    float* s_v = sm + NPAD * 16;       // 208*4
    float* s_a = sm + NPAD * 20;       // 8 waves * 16 * AST

    const int bh = blockIdx.x;
    const int b = bh >> 5, h = bh & 31;
    const float* qg = qb + (size_t)bh * NTOK * 8;   // 6272 B, 16B aligned
    const float* kg = kb + (size_t)bh * NTOK * 8;
    const float* vg = vf + (size_t)bh * NTOK * 4;   // 3136 B, 16B aligned

    const int t = threadIdx.x;
    // zero-fill pad rows (disjoint from async-written region)
    for (int i = NTOK * 8 + t; i < NPAD * 8; i += 256) { s_q[i] = 0.f; s_k[i] = 0.f; }
    for (int i = NTOK * 4 + t; i < NPAD * 4; i += 256) s_v[i] = 0.f;
    // async stage valid region: q/k are 392 x 16B each, v is 196 x 16B
    for (int i = t; i < 392; i += 256) {
        async_copy_b128(&s_q[i * 4], qg + i * 4);
        async_copy_b128(&s_k[i * 4], kg + i * 4);
    }
    for (int i = t; i < 196; i += 256) async_copy_b128(&s_v[i * 4], vg + i * 4);
    wait_asynccnt0();
    __syncthreads();

    const int lane = t & 31, wave = t >> 5;
    const int half = lane >> 4, l16 = lane & 15, koff = half * 2;
    float* aw = s_a + wave * 16 * AST;
    const float* bias_h = bias + (size_t)h * NPAD * NPAD;

    for (int rb = wave; rb < 13; rb += 8) {
        // prefetch next row-block's bias rows (global_prefetch_b8 path)
        if (rb + 8 < 13 && lane < 16)
            __builtin_prefetch(bias_h + (rb + 8) * 16 * NPAD + lane * NPAD, 0, 1);

        const int qrow = rb * 16 + l16;
        v2f a0, a1;
        a0.x = s_q[qrow * 8 + koff];     a0.y = s_q[qrow * 8 + koff + 1];
        a1.x = s_q[qrow * 8 + 4 + koff]; a1.y = s_q[qrow * 8 + 4 + koff + 1];

        v8f c[13];
        #pragma unroll
        for (int j = 0; j < 13; ++j) {
            const int krow = j * 16 + l16;
            v2f b0, b1;
            b0.x = s_k[krow * 8 + koff];     b0.y = s_k[krow * 8 + koff + 1];
            b1.x = s_k[krow * 8 + 4 + koff]; b1.y = s_k[krow * 8 + 4 + koff + 1];
            v8f cc = {};
            cc = wmma_f32_4(a0, b0, cc);     // branch 1 (q pre-scaled)
            cc = wmma_f32_4(a1, b1, cc);     // branch 2
            c[j] = cc;
        }

        // bias add + row max (rows live in 16-lane halves)
        const float* bg = bias_h + rb * 16 * NPAD;
        float rmax[8];
        #pragma unroll
        for (int r = 0; r < 8; ++r) rmax[r] = -1e30f;
        #pragma unroll
        for (int j = 0; j < 13; ++j) {
            #pragma unroll
            for (int r = 0; r < 8; ++r) {
                int m = r + half * 8;
                float v = c[j][r] + bg[m * NPAD + j * 16 + l16];
                c[j][r] = v;
                rmax[r] = fmaxf(rmax[r], v);
            }
        }
        #pragma unroll
        for (int r = 0; r < 8; ++r) {
            float v = rmax[r];
            v = fmaxf(v, __shfl_xor(v, 1, 32));
            v = fmaxf(v, __shfl_xor(v, 2, 32));
            v = fmaxf(v, __shfl_xor(v, 4, 32));
            v = fmaxf(v, __shfl_xor(v, 8, 32));
            rmax[r] = v;
        }
        float rsum[8];
        #pragma unroll
        for (int r = 0; r < 8; ++r) rsum[r] = 0.f;
        #pragma unroll
        for (int j = 0; j < 13; ++j) {
            #pragma unroll
            for (int r = 0; r < 8; ++r) {
                float e = __expf(c[j][r] - rmax[r]);
                c[j][r] = e;
                rsum[r] += e;
            }
        }
        #pragma unroll
        for (int r = 0; r < 8; ++r) {
            float v = rsum[r];
            v += __shfl_xor(v, 1, 32);
            v += __shfl_xor(v, 2, 32);
            v += __shfl_xor(v, 4, 32);
            v += __shfl_xor(v, 8, 32);
            rsum[r] = 1.0f / v;
        }
        // normalized attn -> LDS strip [16][AST]
        #pragma unroll
        for (int j = 0; j < 13; ++j) {
            #pragma unroll
            for (int r = 0; r < 8; ++r) {
                int m = r + half * 8;
                aw[m * AST + j * 16 + l16] = c[j][r] * rsum[r];
            }
        }

        // attn @ v : [16 x 208] x [208 x 4(pad16)] via 52 WMMA k-steps
        v8f co = {};
        #pragma unroll 4
        for (int ks = 0; ks < 52; ++ks) {
            v2f a, bb;
            a.x = aw[l16 * AST + ks * 4 + koff];
            a.y = aw[l16 * AST + ks * 4 + koff + 1];
            int vr = ks * 4 + koff;
            bb.x = (l16 < 4) ? s_v[vr * 4 + l16] : 0.f;
            bb.y = (l16 < 4) ? s_v[(vr + 1) * 4 + l16] : 0.f;
            co = wmma_f32_4(a, bb, co);
        }
        if (l16 < 4) {
            #pragma unroll
            for (int r = 0; r < 8; ++r) {
                int row = rb * 16 + r + half * 8;
                if (row < NTOK)
                    xa[(b * NTOK + row) * EMBEDC + h * 4 + l16] = co[r];
            }
        }
    }
}

// ---------------------------------------------------------------------------
// Kernel 5: output projection GEMM [12544 x 128] @ proj_w^T + proj_b.
// One wave per 16-row tile, 8 N-tiles, 32 K-steps of WMMA f32 16x16x4.
// ---------------------------------------------------------------------------
__global__ __launch_bounds__(32) void proj_kernel(
    const float* __restrict__ xa, const float* __restrict__ pw,
    const float* __restrict__ pb, float* __restrict__ out)
{
    const int mt = blockIdx.x;
    const int lane = threadIdx.x;
    const int half = lane >> 4, l16 = lane & 15, koff = half * 2;
    const float* arow = xa + (mt * 16 + l16) * 128;

    v8f c[8];
    #pragma unroll
    for (int j = 0; j < 8; ++j) c[j] = (v8f){};
    #pragma unroll
    for (int kk = 0; kk < 32; ++kk) {
        v2f a;
        a.x = arow[kk * 4 + koff];
        a.y = arow[kk * 4 + koff + 1];
        #pragma unroll
        for (int j = 0; j < 8; ++j) {
            const float* wr = pw + (j * 16 + l16) * 128;  // B[k,o] = proj_w[o,k]
            v2f bb;
            bb.x = wr[kk * 4 + koff];
            bb.y = wr[kk * 4 + koff + 1];
            c[j] = wmma_f32_4(a, bb, c[j]);
        }
    }
    #pragma unroll
    for (int j = 0; j < 8; ++j) {
        float bias = pb[j * 16 + l16];
        #pragma unroll
        for (int r = 0; r < 8; ++r) {
            int row = mt * 16 + r + half * 8;
            out[row * 128 + j * 16 + l16] = c[j][r] + bias;
        }
    }
}

// ---------------------------------------------------------------------------
extern "C" void kernel_launch(void* const* d_in, const int* in_sizes, int n_in,
                              void* d_out, int out_size, void* d_ws, size_t ws_size,
                              hipStream_t stream) {
    const float* x_in   = (const float*)d_in[0];
    const float* x_in2  = (const float*)d_in[1];
    const float* qkv_w  = (const float*)d_in[2];
    const float* proj_w = (const float*)d_in[3];
    const float* proj_b = (const float*)d_in[4];
    const float* rpb    = (const float*)d_in[5];
    const float* fuse_w = (const float*)d_in[6];
    const float* fuse_b = (const float*)d_in[7];
    const int*   rel    = (const int*)d_in[8];
    float* out = (float*)d_out;

    // Workspace layout (floats)
    float* ws = (float*)d_ws;
    const size_t SZ_QK   = (size_t)BATCH * NHEADS * NTOK * 8;   // 3,211,264
    const size_t SZ_VCAT = (size_t)BATCH * 64 * NTOK * 4;       // 3,211,264
    const size_t SZ_VF   = (size_t)BATCH * 32 * NTOK * 4;       // 1,605,632
    const size_t SZ_BIAS = (size_t)NHEADS * NPAD * NPAD;        // 1,384,448
    float* qb   = ws;
    float* kb   = qb + SZ_QK;
    float* vcat = kb + SZ_QK;
    float* vf   = vcat + SZ_VCAT;
    float* bias = vf + SZ_VF;
    float* xa   = bias + SZ_BIAS;

    const int attn_smem = (NPAD * 20 + 8 * 16 * AST) * sizeof(float); // ~125 KB
    (void)hipFuncSetAttribute((const void*)attn_kernel,
                              hipFuncAttributeMaxDynamicSharedMemorySize,
                              attn_smem);

    qkv_kernel<<<dim3(MTILES, 2), 256, 0, stream>>>(x_in, x_in2, qkv_w, qb, kb, vcat);
    fuse_kernel<<<BATCH, 256, 0, stream>>>(vcat, fuse_w, fuse_b, vf);
    bias_kernel<<<(NHEADS * NPAD * NPAD + 255) / 256, 256, 0, stream>>>(rpb, rel, bias);
    attn_kernel<<<BATCH * NHEADS, 256, attn_smem, stream>>>(qb, kb, vf, bias, xa);
    proj_kernel<<<MTILES, 32, 0, stream>>>(xa, proj_w, proj_b, out);
}